// GINEncoder_48954037240335
// MI455X (gfx1250) — compile-verified
//
#include <hip/hip_runtime.h>

typedef __attribute__((ext_vector_type(2))) float v2f;
typedef __attribute__((ext_vector_type(8))) float v8f;

#define HID 128
#define ROWS 32          // nodes per block in the MLP kernel
#define LDP 132          // padded LDS row stride (floats) to avoid bank conflicts
#define BN_EPS 1e-5f

// ---------------------------------------------------------------- zero
__global__ __launch_bounds__(256) void zero_kernel(float4* __restrict__ p, long long n4) {
  long long i = (long long)blockIdx.x * 256 + threadIdx.x;
  if (i < n4) p[i] = make_float4(0.f, 0.f, 0.f, 0.f);
}

// ---------------------------------------------------------------- weight repack
// wpk[(k/2)*HID + c] = { w[k][c], w[k+1][c] }  -> one b64 load per WMMA B-fragment
__global__ __launch_bounds__(256) void pack_w_kernel(const float* __restrict__ w,
                                                     float2* __restrict__ wpk) {
  int i = blockIdx.x * 256 + threadIdx.x;      // i in [0, HID*HID/2)
  if (i >= HID * HID / 2) return;
  int kp = i >> 7;                              // K-pair index (0..63)
  int c  = i & (HID - 1);
  wpk[i] = make_float2(w[(2 * kp) * HID + c], w[(2 * kp + 1) * HID + c]);
}

// ---------------------------------------------------------------- edge scatter-add
// one wave32 per edge: lane l moves channels [4l, 4l+3] with f32 atomics (L2-resident)
__global__ __launch_bounds__(256) void scatter_kernel(
    const float* __restrict__ x, const int* __restrict__ src,
    const int* __restrict__ dst, float* __restrict__ agg, int nEdges) {
  long long gid = (long long)blockIdx.x * 256 + threadIdx.x;
  int e = (int)(gid >> 5);
  int lane = (int)(gid & 31);
  if (e >= nEdges) return;
  int s = src[e];
  int d = dst[e];
  float4 v = ((const float4*)(x + (size_t)s * HID))[lane];
  float* o = agg + (size_t)d * HID + lane * 4;
  atomicAdd(o + 0, v.x);
  atomicAdd(o + 1, v.y);
  atomicAdd(o + 2, v.z);
  atomicAdd(o + 3, v.w);
}

// ---------------------------------------------------------------- fused GIN MLP
// h = relu((x+agg) @ w1 + b1) @ w2 + b2 ; also accumulates per-channel sum/sumsq
__global__ __launch_bounds__(256) void gin_mlp_kernel(
    const float* __restrict__ xin, const float* __restrict__ agg,
    const float2* __restrict__ w1p, const float* __restrict__ b1,
    const float2* __restrict__ w2p, const float* __restrict__ b2,
    float* __restrict__ hout, float* __restrict__ stats, int nNodes) {
  __shared__ float sh_a[ROWS * LDP];   // (x + agg) tile
  __shared__ float sh_t[ROWS * LDP];   // relu(mlp1) tile
  __shared__ float sh_sum[HID];
  __shared__ float sh_sq[HID];

  const int tid  = threadIdx.x;
  const int row0 = blockIdx.x * ROWS;

  if (tid < HID) {
    sh_sum[tid] = 0.f;
    sh_sq[tid]  = 0.f;
    __builtin_prefetch(w1p + tid * (HID / 2), 0, 0);   // global_prefetch_b8: warm weights
    __builtin_prefetch(w2p + tid * (HID / 2), 0, 0);
  }

  // stage hpre = x + agg into LDS (ROWS x 128, padded stride)
  for (int i = tid; i < ROWS * (HID / 4); i += 256) {
    int r  = i >> 5;      // 32 float4 per row
    int c4 = i & 31;
    int row = row0 + r;
    float4 v = make_float4(0.f, 0.f, 0.f, 0.f);
    if (row < nNodes) {
      float4 a = ((const float4*)(xin + (size_t)row * HID))[c4];
      float4 g = ((const float4*)(agg + (size_t)row * HID))[c4];
      v = make_float4(a.x + g.x, a.y + g.y, a.z + g.z, a.w + g.w);
    }
    float* d = sh_a + r * LDP + c4 * 4;
    d[0] = v.x; d[1] = v.y; d[2] = v.z; d[3] = v.w;
  }
  __syncthreads();

  const int wave = tid >> 5;
  const int lane = tid & 31;
  const int m_t  = wave >> 2;            // 0..1 : which 16-row tile
  const int n0   = (wave & 3) * 32;      // this wave owns 2 column tiles: n0, n0+16
  const int l15  = lane & 15;
  const int hi   = lane >> 4;            // 0 or 1 (half-wave)
  const int kk   = hi * 2;               // A/B fragment K sub-offset
  const bool valid = (row0 + m_t * 16) < nNodes;   // tiles of 16 divide N exactly

  // ---- GEMM1: t = relu(hpre @ w1 + b1), K = 128 as 32 x v_wmma_f32_16x16x4_f32
  {
    v8f acc0 = {};
    v8f acc1 = {};
    if (valid) {
      const float* arow = sh_a + (m_t * 16 + l15) * LDP;
      const int c0 = n0 + l15;
      const int c1 = n0 + 16 + l15;
      for (int k = 0; k < HID; k += 4) {
        v2f a = *(const v2f*)(arow + k + kk);               // A frag: K={k+kk, k+kk+1}
        const float2* wp = w1p + ((k >> 1) + hi) * HID;      // packed K-pair row
        v2f bA = *(const v2f*)(wp + c0);                     // single b64 load
        v2f bB = *(const v2f*)(wp + c1);
        acc0 = __builtin_amdgcn_wmma_f32_16x16x4_f32(false, a, false, bA, (short)0, acc0, false, false);
        acc1 = __builtin_amdgcn_wmma_f32_16x16x4_f32(false, a, false, bB, (short)0, acc1, false, false);
      }
      const float bias0 = b1[c0];
      const float bias1 = b1[c1];
      #pragma unroll
      for (int i = 0; i < 8; ++i) {
        int m = hi * 8 + i;                                 // C/D layout: VGPR i -> M = i + 8*hi
        sh_t[(m_t * 16 + m) * LDP + c0] = fmaxf(acc0[i] + bias0, 0.f);
        sh_t[(m_t * 16 + m) * LDP + c1] = fmaxf(acc1[i] + bias1, 0.f);
      }
    }
  }
  __syncthreads();

  // ---- GEMM2: h = t @ w2 + b2, write out + BN partial sums
  {
    v8f d0 = {};
    v8f d1 = {};
    if (valid) {
      const float* arow = sh_t + (m_t * 16 + l15) * LDP;
      const int c0 = n0 + l15;
      const int c1 = n0 + 16 + l15;
      for (int k = 0; k < HID; k += 4) {
        v2f a = *(const v2f*)(arow + k + kk);
        const float2* wp = w2p + ((k >> 1) + hi) * HID;
        v2f bA = *(const v2f*)(wp + c0);
        v2f bB = *(const v2f*)(wp + c1);
        d0 = __builtin_amdgcn_wmma_f32_16x16x4_f32(false, a, false, bA, (short)0, d0, false, false);
        d1 = __builtin_amdgcn_wmma_f32_16x16x4_f32(false, a, false, bB, (short)0, d1, false, false);
      }
      const float bias0 = b2[c0];
      const float bias1 = b2[c1];
      float s0 = 0.f, q0 = 0.f, s1 = 0.f, q1 = 0.f;
      #pragma unroll
      for (int i = 0; i < 8; ++i) {
        int m = hi * 8 + i;
        size_t row = (size_t)(row0 + m_t * 16 + m);
        float v0 = d0[i] + bias0;
        float v1 = d1[i] + bias1;
        hout[row * HID + c0] = v0;
        hout[row * HID + c1] = v1;
        s0 += v0; q0 += v0 * v0;
        s1 += v1; q1 += v1 * v1;
      }
      atomicAdd(&sh_sum[c0], s0);
      atomicAdd(&sh_sq[c0],  q0);
      atomicAdd(&sh_sum[c1], s1);
      atomicAdd(&sh_sq[c1],  q1);
    }
  }
  __syncthreads();
  if (tid < HID) {
    atomicAdd(&stats[tid],        sh_sum[tid]);
    atomicAdd(&stats[HID + tid],  sh_sq[tid]);
  }
}

// ---------------------------------------------------------------- BN prep: per-channel scale/shift
__global__ __launch_bounds__(HID) void bn_prep_kernel(
    const float* __restrict__ stats, const float* __restrict__ gamma,
    const float* __restrict__ beta, float* __restrict__ ss, int nNodes) {
  int c = threadIdx.x;
  float inv_n = 1.0f / (float)nNodes;
  float mean = stats[c] * inv_n;
  float var  = stats[HID + c] * inv_n - mean * mean;   // biased variance
  float sc   = gamma[c] * rsqrtf(var + BN_EPS);
  ss[c]       = sc;
  ss[HID + c] = beta[c] - mean * sc;
}

// ---------------------------------------------------------------- BN apply + ReLU (streaming)
__global__ __launch_bounds__(256) void bn_relu_kernel(
    const float* __restrict__ h, const float* __restrict__ ss,
    float* __restrict__ out, int nNodes) {
  long long i4 = (long long)blockIdx.x * 256 + threadIdx.x;
  long long n4 = (long long)nNodes * (HID / 4);
  if (i4 >= n4) return;
  int c4 = (int)(i4 & 31);
  float4 hv  = ((const float4*)h)[i4];
  float4 sc4 = ((const float4*)ss)[c4];
  float4 sh4 = ((const float4*)(ss + HID))[c4];
  float4 ov;
  ov.x = fmaxf(hv.x * sc4.x + sh4.x, 0.f);
  ov.y = fmaxf(hv.y * sc4.y + sh4.y, 0.f);
  ov.z = fmaxf(hv.z * sc4.z + sh4.z, 0.f);
  ov.w = fmaxf(hv.w * sc4.w + sh4.w, 0.f);
  ((float4*)out)[i4] = ov;
}

// ---------------------------------------------------------------- launcher
extern "C" void kernel_launch(void* const* d_in, const int* in_sizes, int n_in,
                              void* d_out, int out_size, void* d_ws, size_t ws_size,
                              hipStream_t stream) {
  (void)n_in; (void)out_size; (void)ws_size;
  const float* x    = (const float*)d_in[0];
  const int*   ei   = (const int*)d_in[1];
  const float* w1_0 = (const float*)d_in[2];
  const float* b1_0 = (const float*)d_in[3];
  const float* w2_0 = (const float*)d_in[4];
  const float* b2_0 = (const float*)d_in[5];
  const float* g_0  = (const float*)d_in[6];
  const float* be_0 = (const float*)d_in[7];
  const float* w1_1 = (const float*)d_in[8];
  const float* b1_1 = (const float*)d_in[9];
  const float* w2_1 = (const float*)d_in[10];
  const float* b2_1 = (const float*)d_in[11];
  const float* g_1  = (const float*)d_in[12];
  const float* be_1 = (const float*)d_in[13];

  const int N = in_sizes[0] / HID;
  const int E = in_sizes[1] / 2;
  const int* src = ei;
  const int* dst = ei + E;

  // workspace layout (floats): agg | stats(256) | ss(256) | 4 packed weight mats
  float*  agg   = (float*)d_ws;
  float*  stats = agg + (size_t)N * HID;
  float*  ss    = stats + 2 * HID;
  float2* wpA   = (float2*)(ss + 2 * HID);            // w1_0 packed (HID*HID/2 float2)
  float2* wpB   = wpA + (HID * HID / 2);              // w2_0 packed
  float2* wpC   = wpB + (HID * HID / 2);              // w1_1 packed
  float2* wpD   = wpC + (HID * HID / 2);              // w2_1 packed
  float*  y     = (float*)d_out;                      // layer-0 output staged in d_out
  float*  out   = (float*)d_out;

  const long long zn4 = ((long long)N * HID + 2 * HID) / 4;  // agg + stats contiguous
  const int zb = (int)((zn4 + 255) / 256);
  const int sb = (int)(((long long)E * 32 + 255) / 256);
  const int mb = (N + ROWS - 1) / ROWS;
  const int nb = (int)(((long long)N * (HID / 4) + 255) / 256);
  const int pb = (HID * HID / 2 + 255) / 256;

  // ---- pack weights into WMMA B-fragment layout (once per call)
  pack_w_kernel<<<pb, 256, 0, stream>>>(w1_0, wpA);
  pack_w_kernel<<<pb, 256, 0, stream>>>(w2_0, wpB);
  pack_w_kernel<<<pb, 256, 0, stream>>>(w1_1, wpC);
  pack_w_kernel<<<pb, 256, 0, stream>>>(w2_1, wpD);

  // ---- layer 0
  zero_kernel<<<zb, 256, 0, stream>>>((float4*)agg, zn4);
  scatter_kernel<<<sb, 256, 0, stream>>>(x, src, dst, agg, E);
  gin_mlp_kernel<<<mb, 256, 0, stream>>>(x, agg, wpA, b1_0, wpB, b2_0, agg, stats, N);
  bn_prep_kernel<<<1, HID, 0, stream>>>(stats, g_0, be_0, ss, N);
  bn_relu_kernel<<<nb, 256, 0, stream>>>(agg, ss, y, N);

  // ---- layer 1 (input = y in d_out; final BN overwrites d_out)
  zero_kernel<<<zb, 256, 0, stream>>>((float4*)agg, zn4);
  scatter_kernel<<<sb, 256, 0, stream>>>(y, src, dst, agg, E);
  gin_mlp_kernel<<<mb, 256, 0, stream>>>(y, agg, wpC, b1_1, wpD, b2_1, agg, stats, N);
  bn_prep_kernel<<<1, HID, 0, stream>>>(stats, g_1, be_1, ss, N);
  bn_relu_kernel<<<nb, 256, 0, stream>>>(agg, ss, out, N);
}